// SeparateCodebooksFeatureQuantizerEMA_30932354466465
// MI455X (gfx1250) — compile-verified
//
#include <hip/hip_runtime.h>
#include <hip/hip_bf16.h>
#include <float.h>
#include <math.h>

// ---------------- problem constants ----------------
// features (197,128,768) f32, class_emb (4096,768), feat_emb (4096,768)
// outputs (flat, in order): loss[1], quantized[197*128*3*768], cl_perp[1],
// ft_perp[1], cl_avg[4096], ft_avg[4096], idx[197*128*3], dist[197*128*8192]
constexpr int       kC        = 768;
constexpr int       kK        = 4096;      // per-codebook entries
constexpr int       kRowsCl   = 128;       // class rows (t = 0)
constexpr int       kRowsFt   = 25088;     // feat rows  (t = 1..196)
constexpr int       kRows     = kRowsCl + kRowsFt;   // 25216
constexpr long long kQuantN   = (long long)kRows * 3 * kC;       // 58,097,664
constexpr long long kQuantOff = 1;
constexpr long long kClPerpOff = kQuantOff + kQuantN;            // 58,097,665
constexpr long long kFtPerpOff = kClPerpOff + 1;
constexpr long long kClAvgOff  = kFtPerpOff + 1;
constexpr long long kFtAvgOff  = kClAvgOff + kK;
constexpr long long kIdxOff    = kFtAvgOff + kK;
constexpr long long kDistOff   = kIdxOff + (long long)kRows * 3; // 58,181,507
constexpr int       kDStride   = 2 * kK;   // 8192 padded distance row

typedef __attribute__((ext_vector_type(2))) float        v2f;
typedef __attribute__((ext_vector_type(8))) float        v8f;
typedef __attribute__((ext_vector_type(4))) unsigned int v4u;
typedef __attribute__((ext_vector_type(8))) int          v8i;
typedef __attribute__((ext_vector_type(4))) int          v4i;

#if defined(__AMDGCN__) && __has_builtin(__builtin_amdgcn_wmma_f32_16x16x4_f32)
#define HAVE_WMMA_F32 1
#else
#define HAVE_WMMA_F32 0
#endif

#if defined(__AMDGCN__) && __has_builtin(__builtin_amdgcn_tensor_load_to_lds) && \
    __has_builtin(__builtin_amdgcn_s_wait_tensorcnt)
#define HAVE_TDM 1
#else
#define HAVE_TDM 0
#endif

// ---------------- small utility kernels ----------------
__global__ void zero_kernel(float* p, int n) {
    int i = blockIdx.x * blockDim.x + threadIdx.x;
    if (i < n) p[i] = 0.0f;
}

// one block per row: dst[row] = sum(src[row,:]^2)
__global__ void rownorm_kernel(const float* __restrict__ src, float* __restrict__ dst) {
    __shared__ float s[256];
    const int row = blockIdx.x, t = threadIdx.x;
    const float* p = src + (size_t)row * kC;
    float acc = 0.f;
    for (int c = t; c < kC; c += 256) { float v = p[c]; acc = fmaf(v, v, acc); }
    s[t] = acc; __syncthreads();
    for (int st = 128; st > 0; st >>= 1) { if (t < st) s[t] += s[t + st]; __syncthreads(); }
    if (t == 0) dst[row] = s[0];
}

// fill the "other codebook" half of each padded distance row with FLT_MAX
__global__ void fill_fmax_kernel(float* __restrict__ dist) {
    long long i = (long long)blockIdx.x * blockDim.x + threadIdx.x;
    if (i >= (long long)kRows * kK) return;
    long long row = i >> 12;            // /4096
    int col = (int)(i & (kK - 1));
    long long c = (row < kRowsCl) ? (col + kK) : col;
    dist[row * kDStride + c] = FLT_MAX;
}

// ---------------- Tensor Data Mover tile load ----------------
// D# per CDNA5 ISA: 2D tile of f32, LDS row padding
// 4 DWORDs after every 64 DWORDs (matches LDA = 68 layout below).
#if HAVE_TDM
__device__ __forceinline__ void tdm_load_tile(const float* gsrc, unsigned lds_off,
                                              int rows, int cols, int stride_elems) {
    unsigned long long ga = (unsigned long long)gsrc;
    v4u g0;
    g0[0] = 1u;                                           // count=1 (valid user D#)
    g0[1] = lds_off;                                      // lds_addr (bytes)
    g0[2] = (unsigned)(ga & 0xffffffffu);                 // global_addr[31:0]
    g0[3] = (unsigned)((ga >> 32) & 0x01ffffffu)          // global_addr[56:32]
          | 0x80000000u;                                  // type=2 ("image")
    v8i g1;
    g1[0] = (2 << 16)                                     // data_size = 4 bytes
          | (1 << 20)                                     // pad_enable
          | (5 << 22)                                     // pad_interval: 64 DWORDs
          | (3 << 25);                                    // pad_amount: 4 DWORDs
    g1[1] = (int)((stride_elems & 0xffff) << 16);         // tensor_dim0[15:0]
    g1[2] = (int)(((unsigned)stride_elems >> 16) & 0xffffu)   // tensor_dim0[31:16]
          | (int)((rows & 0xffff) << 16);                 // tensor_dim1[15:0]
    g1[3] = (int)(((unsigned)rows >> 16) & 0xffffu)       // tensor_dim1[31:16]
          | (int)((cols & 0xffff) << 16);                 // tile_dim0
    g1[4] = rows & 0xffff;                                // tile_dim1 (tile_dim2 = 0)
    g1[5] = stride_elems;                                 // tensor_dim0_stride[31:0]
    g1[6] = 0;                                            // stride[47:32], dim1_stride lo
    g1[7] = 0;
    v4i z4 = {0, 0, 0, 0};                                // groups 2/3 unused (2D)
#if __clang_major__ >= 23
    v8i z8 = {0, 0, 0, 0, 0, 0, 0, 0};
    __builtin_amdgcn_tensor_load_to_lds(g0, g1, z4, z4, z8, 0);
#else
    __builtin_amdgcn_tensor_load_to_lds(g0, g1, z4, z4, 0);
#endif
}
#endif

// ---------------- WMMA fp32 distance GEMM ----------------
// D[m,n] = xx[m] + ee[n] - 2 * dot(X[m,:], E[n,:])
// Block tile 64(M) x 32(N), 8 waves (4x2), each wave one 16x16 C tile.
// K staged through LDS in chunks of 64 via TDM (fallback: float4 loads).
#define LDA 68   // 64 + 4 pad floats; row = 272B = 16B multiple

__global__ void __launch_bounds__(256)
dist_gemm_kernel(const float* __restrict__ X, const float* __restrict__ E,
                 const float* __restrict__ xx, const float* __restrict__ ee,
                 float* __restrict__ out, long long out_stride) {
    __shared__ float As[64 * LDA];
    __shared__ float Bs[32 * LDA];

    const int tid  = threadIdx.x;
    const int wave = tid >> 5, lane = tid & 31;
    const int wm = wave & 3, wn = wave >> 2;       // 4 M-waves x 2 N-waves
    const int ln = lane & 15, hi = lane >> 4;      // lane split per WMMA layout
    const int m0 = blockIdx.y * 64, n0 = blockIdx.x * 32;

    v8f c = {};
    for (int kc = 0; kc < kC; kc += 64) {
#if HAVE_TDM
        // wave 0 drives the Tensor Data Mover for both tiles, waits TENSORcnt,
        // then the workgroup barrier releases the compute waves.
        if (wave == 0) {
            tdm_load_tile(X + (size_t)m0 * kC + kc,
                          (unsigned)(unsigned long long)(uintptr_t)As, 64, 64, kC);
            tdm_load_tile(E + (size_t)n0 * kC + kc,
                          (unsigned)(unsigned long long)(uintptr_t)Bs, 32, 64, kC);
            __builtin_amdgcn_s_wait_tensorcnt(0);
        }
#else
        for (int i = tid; i < 64 * 16; i += 256) {
            int r = i >> 4, c4 = (i & 15) << 2;
            float4 v = *reinterpret_cast<const float4*>(X + (size_t)(m0 + r) * kC + kc + c4);
            *reinterpret_cast<float4*>(&As[r * LDA + c4]) = v;
        }
        for (int i = tid; i < 32 * 16; i += 256) {
            int r = i >> 4, c4 = (i & 15) << 2;
            float4 v = *reinterpret_cast<const float4*>(E + (size_t)(n0 + r) * kC + kc + c4);
            *reinterpret_cast<float4*>(&Bs[r * LDA + c4]) = v;
        }
#endif
        if (kc + 64 < kC) {   // global_prefetch_b8 of the next chunk
            __builtin_prefetch(X + (size_t)(m0 + (tid & 63)) * kC + kc + 64, 0, 1);
            __builtin_prefetch(E + (size_t)(n0 + (tid & 31)) * kC + kc + 64, 0, 1);
        }
        __syncthreads();

        const float* Ab = &As[(wm * 16 + ln) * LDA + 2 * hi];
        const float* Bb = &Bs[(wn * 16 + ln) * LDA + 2 * hi];
#pragma unroll
        for (int ks = 0; ks < 64; ks += 4) {
#if HAVE_WMMA_F32
            v2f a = *reinterpret_cast<const v2f*>(Ab + ks);   // A 16x4 frag (2 VGPRs)
            v2f b = *reinterpret_cast<const v2f*>(Bb + ks);   // B 4x16 frag (2 VGPRs)
            c = __builtin_amdgcn_wmma_f32_16x16x4_f32(
                    false, a, false, b, (short)0, c, false, false);
#else
            for (int kk = 0; kk < 4; ++kk) {
#pragma unroll
                for (int r = 0; r < 8; ++r)
                    c[r] = fmaf(As[(wm * 16 + r + 8 * hi) * LDA + ks + kk],
                                Bs[(wn * 16 + ln) * LDA + ks + kk], c[r]);
            }
#endif
        }
        __syncthreads();
    }

    // epilogue: d = xx + ee - 2*dot  (C/D layout: VGPR r -> M = r + 8*hi, N = ln)
    const float eev = ee[n0 + wn * 16 + ln];
#pragma unroll
    for (int r = 0; r < 8; ++r) {
        int gm = m0 + wm * 16 + r + 8 * hi;
        int gn = n0 + wn * 16 + ln;
        out[(long long)gm * out_stride + gn] = fmaf(-2.f, c[r], xx[gm] + eev);
    }
}

// ---------------- top-3 / gather / counts / mse ----------------
__global__ void __launch_bounds__(256)
topk_kernel(const float* __restrict__ dist, const float* __restrict__ features,
            const float* __restrict__ class_emb, const float* __restrict__ feat_emb,
            float* __restrict__ quant_out, float* __restrict__ idx_out,
            float* __restrict__ counts_cl, float* __restrict__ counts_ft,
            float* __restrict__ mse_cl, float* __restrict__ mse_ft) {
    const int row = blockIdx.x;          // 0..25215
    const int t = threadIdx.x;
    const bool isClass = row < kRowsCl;
    const float* drow = dist + (long long)row * kDStride + (isClass ? 0 : kK);
    const float* E = isClass ? class_emb : feat_emb;

    __shared__ float sval[256];
    __shared__ int   sidx[256];
    __shared__ int   chosen[3];

    for (int r = 0; r < 3; ++r) {
        float best = FLT_MAX; int bi = 0x7fffffff;
        for (int k = t; k < kK; k += 256) {
            bool skip = false;
            for (int j = 0; j < r; ++j) skip |= (chosen[j] == k);
            float v = drow[k];
            if (!skip && v < best) { best = v; bi = k; }
        }
        sval[t] = best; sidx[t] = bi; __syncthreads();
        for (int s = 128; s > 0; s >>= 1) {
            if (t < s) {
                if (sval[t + s] < sval[t] ||
                    (sval[t + s] == sval[t] && sidx[t + s] < sidx[t])) {
                    sval[t] = sval[t + s]; sidx[t] = sidx[t + s];
                }
            }
            __syncthreads();
        }
        if (t == 0) chosen[r] = sidx[0];
        __syncthreads();
    }

    // gather quantized rows + commitment-MSE partial
    const float* xrow = features + (size_t)row * kC;
    float acc = 0.f;
    for (int j = 0; j < 3; ++j) {
        const float* erow = E + (size_t)chosen[j] * kC;
        float* qrow = quant_out + ((long long)row * 3 + j) * kC;
        for (int ccol = t; ccol < kC; ccol += 256) {
            float e = erow[ccol];
            qrow[ccol] = e;                   // q_st value == quant in forward
            float d = e - xrow[ccol];
            acc = fmaf(d, d, acc);
        }
    }
    sval[t] = acc; __syncthreads();
    for (int s = 128; s > 0; s >>= 1) { if (t < s) sval[t] += sval[t + s]; __syncthreads(); }
    if (t == 0) atomicAdd(isClass ? mse_cl : mse_ft, sval[0]);
    if (t < 3) {
        atomicAdd((isClass ? counts_cl : counts_ft) + chosen[t], 1.0f);
        idx_out[(long long)row * 3 + t] = (float)(chosen[t] + (isClass ? 0 : kK));
    }
}

// ---------------- finalize: avg_probs, perplexity, loss ----------------
__global__ void finalize_kernel(const float* __restrict__ counts_cl,
                                const float* __restrict__ counts_ft,
                                const float* __restrict__ mse_cl,
                                const float* __restrict__ mse_ft,
                                float* __restrict__ out) {
    __shared__ float s[256];
    const int t = threadIdx.x;
    for (int cb = 0; cb < 2; ++cb) {
        const float* counts = cb ? counts_ft : counts_cl;
        const float n = cb ? (float)kRowsFt : (float)kRowsCl;
        float* avg = out + (cb ? kFtAvgOff : kClAvgOff);
        float acc = 0.f;
        for (int k = t; k < kK; k += 256) {
            float p = counts[k] / n;
            avg[k] = p;
            acc += p * logf(p + 1e-10f);
        }
        s[t] = acc; __syncthreads();
        for (int st = 128; st > 0; st >>= 1) { if (t < st) s[t] += s[t + st]; __syncthreads(); }
        if (t == 0) out[cb ? kFtPerpOff : kClPerpOff] = expf(-s[0]);
        __syncthreads();
    }
    if (t == 0) {
        float clm = *mse_cl / ((float)kRowsCl * 3.f * (float)kC);
        float ftm = *mse_ft / ((float)kRowsFt * 3.f * (float)kC);
        out[0] = 0.25f * ftm + 0.25f * clm;   // ft_loss + cl_loss
    }
}

// ---------------- launcher ----------------
extern "C" void kernel_launch(void* const* d_in, const int* in_sizes, int n_in,
                              void* d_out, int out_size, void* d_ws, size_t ws_size,
                              hipStream_t stream) {
    (void)in_sizes; (void)n_in; (void)out_size; (void)ws_size;
    const float* features  = (const float*)d_in[0];
    const float* class_emb = (const float*)d_in[1];
    const float* feat_emb  = (const float*)d_in[2];
    float* out = (float*)d_out;
    float* ws  = (float*)d_ws;

    // workspace layout (floats)
    float* xsq       = ws;                 // 25216 row norms of features
    float* esq_cl    = ws + 25216;         // 4096
    float* esq_ft    = ws + 29312;         // 4096
    float* counts_cl = ws + 33408;         // 4096
    float* counts_ft = ws + 37504;         // 4096
    float* mse_cl    = ws + 41600;         // 1
    float* mse_ft    = ws + 41601;         // 1

    float* dist  = out + kDistOff;
    float* quant = out + kQuantOff;
    float* idxo  = out + kIdxOff;

    // 1) zero accumulators (counts_cl..mse_ft are contiguous: 8194 floats)
    zero_kernel<<<(8194 + 255) / 256, 256, 0, stream>>>(counts_cl, 8194);

    // 2) row norms
    rownorm_kernel<<<kRows, 256, 0, stream>>>(features, xsq);
    rownorm_kernel<<<kK, 256, 0, stream>>>(class_emb, esq_cl);
    rownorm_kernel<<<kK, 256, 0, stream>>>(feat_emb, esq_ft);

    // 3) FMAX padding of the distance output
    {
        long long n = (long long)kRows * kK;
        fill_fmax_kernel<<<(unsigned)((n + 255) / 256), 256, 0, stream>>>(dist);
    }

    // 4) WMMA fp32 distance GEMMs (write straight into d_out distance region)
    {
        dim3 gcl(kK / 32, kRowsCl / 64);   // (128, 2)
        dist_gemm_kernel<<<gcl, 256, 0, stream>>>(features, class_emb, xsq, esq_cl,
                                                  dist, (long long)kDStride);
        dim3 gft(kK / 32, kRowsFt / 64);   // (128, 392)
        dist_gemm_kernel<<<gft, 256, 0, stream>>>(features + (size_t)kRowsCl * kC, feat_emb,
                                                  xsq + kRowsCl, esq_ft,
                                                  dist + (long long)kRowsCl * kDStride + kK,
                                                  (long long)kDStride);
    }

    // 5) top-3 + gather + counts + mse
    topk_kernel<<<kRows, 256, 0, stream>>>(dist, features, class_emb, feat_emb,
                                           quant, idxo, counts_cl, counts_ft, mse_cl, mse_ft);

    // 6) scalars + avg_probs + perplexity
    finalize_kernel<<<1, 256, 0, stream>>>(counts_cl, counts_ft, mse_cl, mse_ft, out);
}